// SemanticModule_884763263721
// MI455X (gfx1250) — compile-verified
//
#include <hip/hip_runtime.h>

typedef __attribute__((ext_vector_type(2))) float v2f;
typedef __attribute__((ext_vector_type(8))) float v8f;

// ---------------------------------------------------------------------------
// H[N x OUTC] = X[N x K] @ W[K x OUTC] + B[OUTC], for up to three (W,B,H)
// triples selected by blockIdx.y (reads X once per layer).
//
// One wave computes a 16x16 output tile with V_WMMA_F32_16X16X4_F32.
// A fragment (16x4 f32): lanes 0-15  -> row M=lane,    K pair {k0,k0+1}
//                        lanes 16-31 -> row M=lane-16, K pair {k0+2,k0+3}
// B fragment (4x16 f32): lanes 0-15  -> col N=lane,    K pair {k0,k0+1}
//                        lanes 16-31 -> col N=lane-16, K pair {k0+2,k0+3}
// C/D (16x16 f32): VGPR r, lanes 0-15 -> M=r, lanes 16-31 -> M=8+r.
//
// W is staged zero-padded into LDS (KP x ST): the unrolled WMMA loop has no
// guards (padded B rows are 0.0, so padded-K garbage A values contribute
// nothing). A-side indices are clamped with value-selects (no EXEC games);
// garbage only lands in rows/cols that are never stored. Store epilogue has
// a wave-uniform full-tile fast path with straight-line stores.
// ---------------------------------------------------------------------------
template <int K, int OUTC>
__global__ void linear3_wmma(const float* __restrict__ X,
                             const float* __restrict__ W0, const float* __restrict__ B0, float* __restrict__ H0,
                             const float* __restrict__ W1, const float* __restrict__ B1, float* __restrict__ H1,
                             const float* __restrict__ W2, const float* __restrict__ B2, float* __restrict__ H2,
                             int N)
{
    constexpr int NT = (OUTC + 15) / 16;  // output column tiles
    constexpr int ST = NT * 16;           // padded OUTC (LDS stride)
    constexpr int KP = (K + 3) & ~3;      // K padded to multiple of 4

    __shared__ float sW[KP * ST];
    __shared__ float sB[ST];

    const float* W = (blockIdx.y == 0) ? W0 : (blockIdx.y == 1) ? W1 : W2;
    const float* B = (blockIdx.y == 0) ? B0 : (blockIdx.y == 1) ? B1 : B2;
    float*       H = (blockIdx.y == 0) ? H0 : (blockIdx.y == 1) ? H1 : H2;

    const int tid = threadIdx.x;
    for (int i = tid; i < KP * ST; i += (int)blockDim.x) {
        int k = i / ST, c = i - k * ST;
        sW[i] = (k < K && c < OUTC) ? W[k * OUTC + c] : 0.0f;
    }
    for (int i = tid; i < ST; i += (int)blockDim.x)
        sB[i] = (i < OUTC) ? B[i] : 0.0f;
    __syncthreads();

    const int gwave  = (int)((blockIdx.x * blockDim.x + tid) >> 5);
    const int lane   = tid & 31;
    const int mtiles = (N + 15) >> 4;

    if (gwave < mtiles * NT) {            // wave-uniform: EXEC all-ones inside
        const int mtile = gwave / NT;
        const int ntile = gwave - mtile * NT;
        const int m0 = mtile << 4;
        const int n0 = ntile << 4;
        const int half = lane >> 4;       // 0 -> K pair {0,1}; 1 -> {2,3}
        const int l15  = lane & 15;
        const int col  = n0 + l15;        // always < ST

        const int arow  = m0 + l15;
        const int arowc = (arow < N) ? arow : (N - 1);
        const float* xrow = X + (size_t)arowc * K;

        const float bias = sB[col];
        v8f acc;
#pragma unroll
        for (int r = 0; r < 8; ++r) acc[r] = bias;

#pragma unroll
        for (int k0 = 0; k0 < KP; k0 += 4) {
            const int ka  = k0 + half * 2;
            const int kac = (K == KP) ? ka : ((ka <= K - 2) ? ka : (K - 2));
            v2f a, b;
            a.x = xrow[kac];
            a.y = xrow[kac + 1];
            b.x = sW[ka * ST + col];
            b.y = sW[(ka + 1) * ST + col];
            acc = __builtin_amdgcn_wmma_f32_16x16x4_f32(false, a, false, b,
                                                        (short)0, acc, false, false);
        }

        const int mbase = m0 + half * 8;
        float* hcol = H + (size_t)mbase * OUTC + col;
        if (m0 + 16 <= N) {
            // full tile: wave-uniform, straight-line stores (hot path)
            if (col < OUTC) {             // folds away when OUTC == ST
#pragma unroll
                for (int r = 0; r < 8; ++r)
                    hcol[(size_t)r * OUTC] = acc[r];
            }
        } else {
            if (col < OUTC) {
#pragma unroll
                for (int r = 0; r < 8; ++r)
                    if (mbase + r < N) hcol[(size_t)r * OUTC] = acc[r];
            }
        }
    }
}

// ---------------------------------------------------------------------------
// In-degree for mean aggregation: cnt[dst[e]] += 1
// ---------------------------------------------------------------------------
__global__ void degree_f32(const int* __restrict__ dst, float* __restrict__ cnt,
                           int E)
{
    int e = blockIdx.x * blockDim.x + threadIdx.x;
    if (e < E) atomicAdd(&cnt[dst[e]], 1.0f);
}

// ---------------------------------------------------------------------------
// AGG[dst[e]][:] += H[src[e]][:]  (32 channels; 8 lanes per edge, float4 each)
// ---------------------------------------------------------------------------
__global__ void scatter_add32(const float* __restrict__ H,
                              const int* __restrict__ src,
                              const int* __restrict__ dst,
                              float* __restrict__ AGG, int E)
{
    int t = blockIdx.x * blockDim.x + threadIdx.x;
    int e = t >> 3;
    if (e >= E) return;
    int g = (t & 7) << 2;                 // channel group start: 0,4,...,28
    int s = src[e];
    int d = dst[e];
    float4 v = *reinterpret_cast<const float4*>(H + (size_t)s * 32 + g);
    float* a = AGG + (size_t)d * 32 + g;
    atomicAdd(a + 0, v.x);
    atomicAdd(a + 1, v.y);
    atomicAdd(a + 2, v.z);
    atomicAdd(a + 3, v.w);
}

// ---------------------------------------------------------------------------
// out = relu(aggt + aggi / max(cnt,1))   (aggt already contains the residual)
// float4 vectorized: one thread per 4 channels.
// ---------------------------------------------------------------------------
__global__ void combine_relu4(const float* __restrict__ aggt,
                              const float* __restrict__ aggi,
                              const float* __restrict__ cnt,
                              float* __restrict__ out, int N)
{
    int idx = blockIdx.x * blockDim.x + threadIdx.x;   // over N*8 float4 groups
    if (idx >= N * 8) return;
    int n = idx >> 3;
    float inv = 1.0f / fmaxf(cnt[n], 1.0f);
    float4 t = reinterpret_cast<const float4*>(aggt)[idx];
    float4 i = reinterpret_cast<const float4*>(aggi)[idx];
    float4 o;
    o.x = fmaxf(t.x + i.x * inv, 0.0f);
    o.y = fmaxf(t.y + i.y * inv, 0.0f);
    o.z = fmaxf(t.z + i.z * inv, 0.0f);
    o.w = fmaxf(t.w + i.w * inv, 0.0f);
    reinterpret_cast<float4*>(out)[idx] = o;
}

// ---------------------------------------------------------------------------
extern "C" void kernel_launch(void* const* d_in, const int* in_sizes, int n_in,
                              void* d_out, int out_size, void* d_ws, size_t ws_size,
                              hipStream_t stream)
{
    const float* x   = (const float*)d_in[0];
    const int*  ei_t = (const int*)d_in[1];
    const int*  ei_i = (const int*)d_in[2];
    const float* W1t = (const float*)d_in[3],  *b1t = (const float*)d_in[4];
    const float* W1i = (const float*)d_in[5],  *b1i = (const float*)d_in[6];
    const float* W1r = (const float*)d_in[7],  *b1r = (const float*)d_in[8];
    const float* W2t = (const float*)d_in[9],  *b2t = (const float*)d_in[10];
    const float* W2i = (const float*)d_in[11], *b2i = (const float*)d_in[12];
    const float* W2r = (const float*)d_in[13], *b2r = (const float*)d_in[14];
    const float* Wc  = (const float*)d_in[15], *bc  = (const float*)d_in[16];
    float* out = (float*)d_out;

    const int N  = in_sizes[0] / 6;
    const int Et = in_sizes[1] / 2;
    const int Ei = in_sizes[2] / 2;
    const int* srcT = ei_t;       const int* dstT = ei_t + Et;
    const int* srcI = ei_i;       const int* dstI = ei_i + Ei;

    // Workspace layout (floats): 5*N*32 + N  (~64.4 MB for N=100k)
    float* ws = (float*)d_ws;
    const size_t F = (size_t)N * 32;
    float* ht   = ws;            // transformed (temp path)
    float* hi   = ws + 1 * F;    // transformed (inter path)
    float* aggt = ws + 2 * F;    // residual + sum aggregation
    float* aggi = ws + 3 * F;    // sum for mean aggregation
    float* h    = ws + 4 * F;    // layer output
    float* cnt  = ws + 5 * F;    // in-degree (inter), N floats

    const int BS = 256;
    auto cdiv = [](long long a, long long b) { return (int)((a + b - 1) / b); };

    const int mtiles = (N + 15) / 16;
    const int linB32 = cdiv((long long)mtiles * 2 * 32, BS); // OUTC=32 -> 2 ntiles
    const int linB7  = cdiv((long long)mtiles * 1 * 32, BS); // OUTC=7  -> 1 ntile
    const int scatBT = cdiv((long long)Et * 8, BS);
    const int scatBI = cdiv((long long)Ei * 8, BS);
    const int combB  = cdiv((long long)N * 8, BS);

    // ---- degree (shared by both layers: same inter edge list) ----
    hipMemsetAsync(cnt, 0, (size_t)N * sizeof(float), stream);
    degree_f32<<<cdiv(Ei, BS), BS, 0, stream>>>(dstI, cnt, Ei);

    // ================= Layer 1 (K = 6) =================
    hipMemsetAsync(aggi, 0, F * sizeof(float), stream);
    linear3_wmma<6, 32><<<dim3(linB32, 3), BS, 0, stream>>>(
        x, W1t, b1t, ht, W1i, b1i, hi, W1r, b1r, aggt, N);
    scatter_add32<<<scatBT, BS, 0, stream>>>(ht, srcT, dstT, aggt, Et);
    scatter_add32<<<scatBI, BS, 0, stream>>>(hi, srcI, dstI, aggi, Ei);
    combine_relu4<<<combB, BS, 0, stream>>>(aggt, aggi, cnt, h, N);

    // ================= Layer 2 (K = 32) =================
    hipMemsetAsync(aggi, 0, F * sizeof(float), stream);
    linear3_wmma<32, 32><<<dim3(linB32, 3), BS, 0, stream>>>(
        h, W2t, b2t, ht, W2i, b2i, hi, W2r, b2r, aggt, N);
    scatter_add32<<<scatBT, BS, 0, stream>>>(ht, srcT, dstT, aggt, Et);
    scatter_add32<<<scatBI, BS, 0, stream>>>(hi, srcI, dstI, aggi, Ei);
    combine_relu4<<<combB, BS, 0, stream>>>(aggt, aggi, cnt, h, N);

    // ================= Classifier (32 -> 7) =================
    linear3_wmma<32, 7><<<dim3(linB7, 1), BS, 0, stream>>>(
        h, Wc, bc, out, Wc, bc, out, Wc, bc, out, N);

    (void)n_in; (void)out_size; (void)ws_size;
}